// LLFullObjectCondensation_5428838662736
// MI455X (gfx1250) — compile-verified
//
#include <hip/hip_runtime.h>
#include <hip/hip_bf16.h>
#include <math.h>

typedef __attribute__((ext_vector_type(2))) float v2f;
typedef __attribute__((ext_vector_type(8))) float v8f;

#define KOBJ 64

// ---------------- helpers ----------------
__device__ __forceinline__ float clampb(float x) {
    return fminf(fmaxf(x, 1e-6f), 1.0f - 1e-6f);
}
__device__ __forceinline__ float huber_f(float x, float d) {
    float ax = fabsf(x);
    return (ax < d) ? x * x : (d * d + 2.0f * d * (ax - d));
}
__device__ __forceinline__ float softclip_f(float x, float s) {
    float y = x / s;
    y = (y > 1.0f) ? logf(y + 1.0f) : y;
    return y * s;
}
// fast q(b) = atanh(b)^2 + 0.1 : v_rcp_f32 + v_log_f32 (TRANS ops)
__device__ __forceinline__ float q_of_b(float b) {
    float at = 0.5f * __logf((1.0f + b) * __builtin_amdgcn_rcpf(1.0f - b));
    return at * at + 0.1f;
}

// ws float-area layout (after 64 u64 packed argmax slots = 512 bytes)
#define F_ATT    0
#define F_REP    64
#define F_XAX    128
#define F_XAY    192
#define F_QAK    256
#define F_BAK    320
#define F_VALID  384
#define F_ACC    448   // pw_den, num_e, num_p, num_t, num_c (5 x 64)
#define F_SC     768   // n_noise_sum, sum_b_noise
#define F_TOTAL  772

// ---------------- kernel 0: zero-init workspace ----------------
__global__ void oc_init_kernel(unsigned long long* packed, float* fbase) {
    int t = threadIdx.x;
    if (t < KOBJ) packed[t] = 0ULL;
    for (int j = t; j < F_TOTAL; j += blockDim.x) fbase[j] = 0.0f;
}

// ---------------- kernel 1: noise stats + per-object argmax ----------------
__global__ void oc_stats_kernel(const float* __restrict__ beta,
                                const int* __restrict__ t_idx,
                                unsigned long long* __restrict__ packed,
                                float* __restrict__ fbase, int n) {
    int i = blockIdx.x * blockDim.x + threadIdx.x;
    float noise = 0.0f, bn = 0.0f;
    if (i < n) {
        int tid = t_idx[i];
        float b = clampb(beta[i]);
        if (tid == 0) {
            noise = 1.0f; bn = b;
        } else {
            unsigned long long p =
                ((unsigned long long)__float_as_uint(b) << 32) | (unsigned int)i;
            atomicMax(&packed[tid - 1], p);
        }
    }
    for (int o = 16; o > 0; o >>= 1) {
        noise += __shfl_xor(noise, o, 32);
        bn    += __shfl_xor(bn, o, 32);
    }
    if ((threadIdx.x & 31) == 0) {
        atomicAdd(&fbase[F_SC + 0], noise);
        atomicAdd(&fbase[F_SC + 1], bn);
    }
}

// ---------------- kernel 2: gather alpha-hit data per object ----------------
__global__ void oc_alpha_kernel(const float* __restrict__ beta,
                                const float* __restrict__ ccoords,
                                const unsigned long long* __restrict__ packed,
                                float* __restrict__ fbase) {
    int k = threadIdx.x;
    if (k >= KOBJ) return;
    unsigned long long p = packed[k];
    if (p == 0ULL) {
        fbase[F_XAX + k] = 0.0f;  fbase[F_XAY + k] = 0.0f;
        fbase[F_QAK + k] = 0.0f;
        fbase[F_BAK + k] = 1.0f;  fbase[F_VALID + k] = 0.0f;
    } else {
        int alpha = (int)(p & 0xffffffffULL);
        float x = ccoords[2 * alpha], y = ccoords[2 * alpha + 1];
        float b = clampb(beta[alpha]);
        float at = atanhf(b);
        fbase[F_XAX + k] = x;  fbase[F_XAY + k] = y;
        fbase[F_QAK + k] = at * at + 0.1f;
        fbase[F_BAK + k] = b;  fbase[F_VALID + k] = 1.0f;
    }
}

// ---------------- kernel 3: all-pairs rep term via WMMA f32 16x16x4 --------
// d2 is produced DIRECTLY by the WMMA by packing norms into the K=2,3 slots:
//   A row i (lanes 0-15):  (x_i.x, x_i.y)   A lanes 16-31 (K=2,3): (|x_i|^2, 1)
//   B col k (lanes 0-15):  (-2*x_k.x, -2*x_k.y)  B lanes 16-31: (1, |x_k|^2)
//   => D[i][k] = |x_i|^2 + |x_k|^2 - 2 x_i.x_k = d2(i,k)
// This kernel accumulates rep_all_k = q_k * sum_i q_i * relu(1 - d(i,k));
// the assigned-pair exclusion is subtracted exactly in the payload kernel.
__global__ void oc_rep_kernel(const float* __restrict__ beta,
                              const float* __restrict__ ccoords,
                              const float* __restrict__ fbase,
                              float* __restrict__ fout,
                              int numTiles) {
    const int lane = threadIdx.x & 31;
    const int wavesPerBlock = blockDim.x >> 5;
    const int waveId = blockIdx.x * wavesPerBlock + (threadIdx.x >> 5);
    const int waveStride = gridDim.x * wavesPerBlock;
    const int col = lane & 15;
    const bool lo = (lane < 16);
    const int rbase = lo ? 0 : 8;

    v2f bfr[4];
    float qk[4];
#pragma unroll
    for (int kt = 0; kt < 4; ++kt) {
        int kk = kt * 16 + col;
        float bx = fbase[F_XAX + kk];
        float by = fbase[F_XAY + kk];
        bfr[kt].x = lo ? (-2.0f * bx) : 1.0f;
        bfr[kt].y = lo ? (-2.0f * by) : (bx * bx + by * by);
        qk[kt] = fbase[F_QAK + kk];
    }
    float acc[4] = {0.f, 0.f, 0.f, 0.f};
    v8f c0 = {};

    for (int tile = waveId; tile < numTiles; tile += waveStride) {
        int hi = tile * 16 + col;
        float ax = ccoords[2 * hi];
        float ay = ccoords[2 * hi + 1];
        float q_l = q_of_b(clampb(beta[hi]));
        float norm_l = ax * ax + ay * ay;

        v2f a;
        a.x = lo ? ax : norm_l;
        a.y = lo ? ay : 1.0f;

        float qr[8];
#pragma unroll
        for (int r = 0; r < 8; ++r)
            qr[r] = __shfl(q_l, rbase + r, 32);

#pragma unroll
        for (int kt = 0; kt < 4; ++kt) {
            v8f d = __builtin_amdgcn_wmma_f32_16x16x4_f32(
                false, a, false, bfr[kt], (short)0, c0, false, false);
#pragma unroll
            for (int r = 0; r < 8; ++r) {
                // v_med3 clamp, v_sqrt (TRANS), v_med3 relu, v_fmac
                float d2c = __builtin_amdgcn_fmed3f(d[r], 0.0f, 3.0e38f);
                float dist = __builtin_amdgcn_sqrtf(d2c + 1e-9f);
                float w = __builtin_amdgcn_fmed3f(1.0f - dist, 0.0f, 1.0f);
                acc[kt] += qr[r] * w;
            }
        }
    }
#pragma unroll
    for (int kt = 0; kt < 4; ++kt)
        atomicAdd(&fout[F_REP + kt * 16 + col], acc[kt] * qk[kt]);
}

// ---------------- kernel 3b: scalar tail (n not multiple of 16) ------------
__global__ void oc_rep_tail_kernel(const float* __restrict__ beta,
                                   const float* __restrict__ ccoords,
                                   const float* __restrict__ fbase,
                                   float* __restrict__ fout,
                                   int start, int n) {
    int i = start + threadIdx.x;
    if (i >= n) return;
    float x = ccoords[2 * i], y = ccoords[2 * i + 1];
    float q = q_of_b(clampb(beta[i]));
    for (int k = 0; k < KOBJ; ++k) {
        float dx = x - fbase[F_XAX + k];
        float dy = y - fbase[F_XAY + k];
        float dist = __builtin_amdgcn_sqrtf(dx * dx + dy * dy + 1e-9f);
        float w = __builtin_amdgcn_fmed3f(1.0f - dist, 0.0f, 1.0f);
        atomicAdd(&fout[F_REP + k], q * fbase[F_QAK + k] * w);
    }
}

// ---------------- kernel 4: per-hit terms, scatter by t_idx ----------------
// LDS slots: 0=pw_den, 1=num_e, 2=num_p, 3=num_t, 4=num_c, 5=att, 6=rep_corr
__global__ void oc_payload_kernel(const float* __restrict__ beta,
                                  const float* __restrict__ ccoords,
                                  const float* __restrict__ pred_energy,
                                  const float* __restrict__ pred_pos,
                                  const float* __restrict__ pred_time,
                                  const float* __restrict__ pred_id,
                                  const float* __restrict__ t_energy,
                                  const float* __restrict__ t_pos,
                                  const float* __restrict__ t_time,
                                  const int* __restrict__ t_idx,
                                  float* __restrict__ fbase, int n) {
    __shared__ float s[7 * KOBJ];
    for (int j = threadIdx.x; j < 7 * KOBJ; j += blockDim.x) s[j] = 0.0f;
    __syncthreads();
    for (int i = blockIdx.x * blockDim.x + threadIdx.x; i < n;
         i += gridDim.x * blockDim.x) {
        int tid = t_idx[i];
        if (tid <= 0) continue;
        int o = tid - 1;
        float b = clampb(beta[i]);
        // payload losses
        float te = t_energy[i];
        float we = (te > 10.0f) ? 1.0f : (te - 0.5f) / 9.5f;
        we = fmaxf(we, 0.0f);
        float pe = pred_energy[i];
        float den = te + 1.0f;
        float el = (den == 0.0f) ? 0.0f : (te - pe) * (te - pe) / den;
        el = softclip_f(el, 10.0f);
        float dx = t_pos[2 * i] - pred_pos[2 * i];
        float dy = t_pos[2 * i + 1] - pred_pos[2 * i + 1];
        float dp2 = dx * dx + dy * dy;
        float pl = softclip_f(huber_f(sqrtf(dp2 / 100.0f + 0.01f), 10.0f), 3.0f);
        float tl = softclip_f(huber_f(t_time[i] - pred_time[i], 2.0f), 6.0f);
        float cl = 0.0f;
        for (int c = 0; c < 6; ++c) { float v = pred_id[6 * i + c]; cl += v * v; }
        cl *= (1e-8f / 6.0f);
        float w = b * we;
        // attractive term + rep exclusion for this hit's own object (exact)
        float cx = ccoords[2 * i] - fbase[F_XAX + o];
        float cy = ccoords[2 * i + 1] - fbase[F_XAY + o];
        float d2 = cx * cx + cy * cy;
        float dist = sqrtf(d2 + 1e-9f);
        float at = atanhf(b);
        float qq = (at * at + 0.1f) * fbase[F_QAK + o];
        atomicAdd(&s[0 * KOBJ + o], b);
        atomicAdd(&s[1 * KOBJ + o], w * el);
        atomicAdd(&s[2 * KOBJ + o], w * pl);
        atomicAdd(&s[3 * KOBJ + o], w * tl);
        atomicAdd(&s[4 * KOBJ + o], w * cl);
        atomicAdd(&s[5 * KOBJ + o], qq * d2);
        atomicAdd(&s[6 * KOBJ + o], qq * fmaxf(1.0f - dist, 0.0f));
    }
    __syncthreads();
    for (int j = threadIdx.x; j < 5 * KOBJ; j += blockDim.x)
        atomicAdd(&fbase[F_ACC + j], s[j]);
    for (int k = threadIdx.x; k < KOBJ; k += blockDim.x) {
        atomicAdd(&fbase[F_ATT + k], s[5 * KOBJ + k]);
        atomicAdd(&fbase[F_REP + k], -s[6 * KOBJ + k]);
    }
}

// ---------------- kernel 5: final combine (one wave) -----------------------
__global__ void oc_final_kernel(const float* __restrict__ fbase,
                                float* __restrict__ out, float Nf) {
    int l = threadIdx.x; // 32 threads
    float vsum = 0.f, att = 0.f, rep = 0.f, bobj = 0.f;
    float le = 0.f, lp = 0.f, lt = 0.f, lc = 0.f;
    for (int k = l; k < KOBJ; k += 32) {
        float v = fbase[F_VALID + k];
        vsum += v;
        att  += fbase[F_ATT + k] * v;
        rep  += fbase[F_REP + k] * v;
        bobj += (1.0f - fbase[F_BAK + k]) * v;
        float den = fbase[F_ACC + 0 * KOBJ + k] + 1e-9f;
        le += fbase[F_ACC + 1 * KOBJ + k] / den * v;
        lp += fbase[F_ACC + 2 * KOBJ + k] / den * v;
        lt += fbase[F_ACC + 3 * KOBJ + k] / den * v;
        lc += fbase[F_ACC + 4 * KOBJ + k] / den * v;
    }
    for (int o = 16; o > 0; o >>= 1) {
        vsum += __shfl_xor(vsum, o, 32);
        att  += __shfl_xor(att, o, 32);
        rep  += __shfl_xor(rep, o, 32);
        bobj += __shfl_xor(bobj, o, 32);
        le   += __shfl_xor(le, o, 32);
        lp   += __shfl_xor(lp, o, 32);
        lt   += __shfl_xor(lt, o, 32);
        lc   += __shfl_xor(lc, o, 32);
    }
    if (l == 0) {
        float nv = fmaxf(vsum, 1.0f);
        float nn = fmaxf(fbase[F_SC + 0], 1.0f);
        float L_att  = att / Nf / nv;
        float L_rep  = rep / Nf / nv;
        float L_beta = bobj / nv + fbase[F_SC + 1] / nn; // S_B = 1
        out[0] = L_att + L_rep + L_beta + (le + lp + lt + lc) / nv;
    }
}

// ---------------- launch -----------------------------------------------
extern "C" void kernel_launch(void* const* d_in, const int* in_sizes, int n_in,
                              void* d_out, int out_size, void* d_ws, size_t ws_size,
                              hipStream_t stream) {
    const float* beta        = (const float*)d_in[0];
    const float* ccoords     = (const float*)d_in[1];
    const float* pred_energy = (const float*)d_in[2];
    const float* pred_pos    = (const float*)d_in[3];
    const float* pred_time   = (const float*)d_in[4];
    const float* pred_id     = (const float*)d_in[5];
    const float* t_energy    = (const float*)d_in[6];
    const float* t_pos       = (const float*)d_in[7];
    const float* t_time      = (const float*)d_in[8];
    const int*   t_idx       = (const int*)d_in[9];
    float* out = (float*)d_out;
    int n = in_sizes[0];

    unsigned long long* packed = (unsigned long long*)d_ws;
    float* fbase = (float*)((char*)d_ws + KOBJ * sizeof(unsigned long long));

    oc_init_kernel<<<1, 256, 0, stream>>>(packed, fbase);

    int blocks1 = (n + 255) / 256;
    oc_stats_kernel<<<blocks1, 256, 0, stream>>>(beta, t_idx, packed, fbase, n);

    oc_alpha_kernel<<<1, KOBJ, 0, stream>>>(beta, ccoords, packed, fbase);

    int numTiles = n / 16;
    if (numTiles > 0)
        oc_rep_kernel<<<128, 256, 0, stream>>>(beta, ccoords, fbase, fbase,
                                               numTiles);
    if (n % 16)
        oc_rep_tail_kernel<<<1, 32, 0, stream>>>(beta, ccoords, fbase, fbase,
                                                 numTiles * 16, n);

    oc_payload_kernel<<<256, 256, 0, stream>>>(beta, ccoords, pred_energy,
                                               pred_pos, pred_time, pred_id,
                                               t_energy, t_pos, t_time, t_idx,
                                               fbase, n);

    oc_final_kernel<<<1, 32, 0, stream>>>(fbase, out, (float)n);
}